// MultiHeadedAttention_65317862637821
// MI455X (gfx1250) — compile-verified
//
#include <hip/hip_runtime.h>

// MHA forward for MI455X (gfx1250, wave32).
// fp32 throughout via V_WMMA_F32_16X16X4_F32 (native fp32 matrix path).
// Attention stripe (32 rows x 2048 cols) lives in ~256KB LDS so the 1.07GB
// attn_weights matrix is written to HBM exactly once and never re-read.
// V is produced in transposed (B,H,D,S) layout so every GEMM fragment load
// is a contiguous float2 (global_load_b64 / ds_load_b64).

typedef __attribute__((ext_vector_type(2))) float v2f;
typedef __attribute__((ext_vector_type(4))) float v4f;
typedef __attribute__((ext_vector_type(8))) float v8f;

#define B_  4
#define S_  2048
#define DM_ 1024
#define H_  16
#define D_  64
#define LDSTRIDE 2052   // pad: 16B row alignment + conflict-free col-pair reads

__device__ __forceinline__ v8f wmma_f32(v2f a, v2f b, v8f c) {
    return __builtin_amdgcn_wmma_f32_16x16x4_f32(
        /*neg_a=*/false, a, /*neg_b=*/false, b,
        /*c_mod=*/(short)0, c, /*reuse_a=*/false, /*reuse_b=*/false);
}

// Y = X @ W^T + bias ; optional zeroing by per-(b,s) mask.
// layout: 0 -> flat (B,S,DM); 1 -> (B,H,S,D); 2 -> transposed (B,H,D,S).
// Block: 256 thr = 8 waves; block tile 128(M) x 64(N); wave tile 16 x 64.
__global__ __launch_bounds__(256)
void proj_kernel(const float* __restrict__ X, const float* __restrict__ W,
                 const float* __restrict__ bias, const unsigned char* __restrict__ mask,
                 float* __restrict__ out, int layout)
{
    const int lane = threadIdx.x & 31;
    const int wave = threadIdx.x >> 5;
    const int half = lane >> 4;
    const int lm   = lane & 15;

    const int tileN = blockIdx.x * 64;
    const int rowM  = blockIdx.y * 128 + wave * 16;   // 16 rows for this wave

    const float* Arow = X + (size_t)(rowM + lm) * DM_ + 2 * half;
    const float* W0 = W + (size_t)(tileN +  0 + lm) * DM_ + 2 * half;
    const float* W1 = W + (size_t)(tileN + 16 + lm) * DM_ + 2 * half;
    const float* W2 = W + (size_t)(tileN + 32 + lm) * DM_ + 2 * half;
    const float* W3 = W + (size_t)(tileN + 48 + lm) * DM_ + 2 * half;

    v8f acc0 = {}, acc1 = {}, acc2 = {}, acc3 = {};

    // register double-buffer: prefetch k-step kb+4 while WMMAs consume kb
    v2f a_c  = *(const v2f*)(Arow);
    v2f b0_c = *(const v2f*)(W0);
    v2f b1_c = *(const v2f*)(W1);
    v2f b2_c = *(const v2f*)(W2);
    v2f b3_c = *(const v2f*)(W3);
    #pragma unroll 4
    for (int kb = 0; kb < DM_ - 4; kb += 4) {
        v2f a_n  = *(const v2f*)(Arow + kb + 4);
        v2f b0_n = *(const v2f*)(W0 + kb + 4);
        v2f b1_n = *(const v2f*)(W1 + kb + 4);
        v2f b2_n = *(const v2f*)(W2 + kb + 4);
        v2f b3_n = *(const v2f*)(W3 + kb + 4);
        acc0 = wmma_f32(a_c, b0_c, acc0);
        acc1 = wmma_f32(a_c, b1_c, acc1);
        acc2 = wmma_f32(a_c, b2_c, acc2);
        acc3 = wmma_f32(a_c, b3_c, acc3);
        a_c = a_n; b0_c = b0_n; b1_c = b1_n; b2_c = b2_n; b3_c = b3_n;
    }
    acc0 = wmma_f32(a_c, b0_c, acc0);
    acc1 = wmma_f32(a_c, b1_c, acc1);
    acc2 = wmma_f32(a_c, b2_c, acc2);
    acc3 = wmma_f32(a_c, b3_c, acc3);

    v8f accs[4] = {acc0, acc1, acc2, acc3};
    #pragma unroll
    for (int nt = 0; nt < 4; ++nt) {
        const int n = tileN + nt * 16 + lm;
        const float bv = bias[n];
        #pragma unroll
        for (int r = 0; r < 8; ++r) {
            const int m = rowM + r + 8 * half;     // global row (b*S + s)
            float y = accs[nt][r] + bv;
            if (mask[m]) y = 0.0f;
            const int b = m >> 11;                 // /S_
            const int s = m & (S_ - 1);
            const int h = n >> 6;                  // /D_
            const int d = n & (D_ - 1);
            if (layout == 1) {
                out[(((size_t)b * H_ + h) * S_ + s) * D_ + d] = y;
            } else if (layout == 2) {
                out[(((size_t)b * H_ + h) * D_ + d) * S_ + s] = y;
            } else {
                out[(size_t)m * DM_ + n] = y;
            }
        }
    }
}

// Fused scores -> softmax -> probs@V for one (b,h) and 32 query rows.
// Q,K in (B,H,S,D); Vt in (B,H,D,S). LDS holds full 32 x 2048 prob stripe.
__global__ __launch_bounds__(256)
void attn_kernel(const float* __restrict__ Q, const float* __restrict__ K,
                 const float* __restrict__ Vt, const unsigned char* __restrict__ kv_mask,
                 float* __restrict__ probs, float* __restrict__ ctx)
{
    extern __shared__ float lds[];                 // 32 * LDSTRIDE floats
    const int lane = threadIdx.x & 31;
    const int wave = threadIdx.x >> 5;
    const int half = lane >> 4;
    const int lm   = lane & 15;

    const int bh    = blockIdx.x;                  // 0..B*H-1
    const int b     = bh >> 4;                     // /H_
    const int h     = bh & (H_ - 1);
    const int qbase = blockIdx.y * 32;

    const float* Qb  = Q  + (size_t)bh * S_ * D_;
    const float* Kb  = K  + (size_t)bh * S_ * D_;
    const float* Vtb = Vt + (size_t)bh * D_ * S_;

    // ---- Phase 1: scores = Q K^T / sqrt(D), masked, into LDS ----
    const int rt = wave >> 2;                      // 0/1: 16-row subtile
    const float* Aq = Qb + (size_t)(qbase + rt * 16 + lm) * D_ + 2 * half;

    // hoist the wave's Q fragment: 16 x v2f covers all of K-dim (D=64)
    v2f afrag[16];
    #pragma unroll
    for (int i = 0; i < 16; ++i) afrag[i] = *(const v2f*)(Aq + 4 * i);

    for (int ct = (wave & 3); ct < S_ / 16; ct += 4) {
        const float* Bk = Kb + (size_t)(ct * 16 + lm) * D_ + 2 * half;
        v8f acc = {};
        #pragma unroll
        for (int i = 0; i < 16; ++i) {
            v2f bf = *(const v2f*)(Bk + 4 * i);
            acc = wmma_f32(afrag[i], bf, acc);
        }
        const int col = ct * 16 + lm;
        const bool km = kv_mask[b * S_ + col] != 0;
        #pragma unroll
        for (int r = 0; r < 8; ++r) {
            float sc = acc[r] * 0.125f;            // 1/sqrt(64)
            if (km) sc = -__builtin_inff();
            lds[(rt * 16 + r + 8 * half) * LDSTRIDE + col] = sc;
        }
    }
    __syncthreads();

    // ---- Phase 2: row softmax (wave per row), float4-vectorized ----
    #pragma unroll
    for (int rr = 0; rr < 4; ++rr) {
        const int row = wave * 4 + rr;
        float* rowp = lds + row * LDSTRIDE;
        float mx = -__builtin_inff();
        for (int j = lane * 4; j < S_; j += 128) {
            v4f v = *(const v4f*)(rowp + j);
            mx = fmaxf(mx, fmaxf(fmaxf(v.x, v.y), fmaxf(v.z, v.w)));
        }
        #pragma unroll
        for (int off = 16; off; off >>= 1) mx = fmaxf(mx, __shfl_xor(mx, off, 32));
        float sum = 0.0f;
        for (int j = lane * 4; j < S_; j += 128) {
            v4f v = *(const v4f*)(rowp + j);
            v.x = __expf(v.x - mx); v.y = __expf(v.y - mx);
            v.z = __expf(v.z - mx); v.w = __expf(v.w - mx);
            *(v4f*)(rowp + j) = v;
            sum += (v.x + v.y) + (v.z + v.w);
        }
        #pragma unroll
        for (int off = 16; off; off >>= 1) sum += __shfl_xor(sum, off, 32);
        const float inv = 1.0f / sum;
        float* gout = probs + ((size_t)bh * S_ + qbase + row) * S_;
        for (int j = lane * 4; j < S_; j += 128) {
            v4f v = *(const v4f*)(rowp + j);
            v.x *= inv; v.y *= inv; v.z *= inv; v.w *= inv;
            *(v4f*)(rowp + j) = v;
            *(v4f*)(gout + j) = v;                 // coalesced b128, written once
        }
    }
    __syncthreads();

    // ---- Phase 3: ctx(32x64) = probs(32x2048) @ V(2048x64); 1 tile/wave ----
    {
        const int rt3 = wave >> 2;
        const int ctn = (wave & 3) * 16;
        const float* Alds = lds + (rt3 * 16 + lm) * LDSTRIDE + 2 * half;
        const float* Bv   = Vtb + (size_t)(ctn + lm) * S_ + 2 * half; // V^T row
        v8f acc = {};
        v2f a_c = *(const v2f*)(Alds);
        v2f b_c = *(const v2f*)(Bv);
        #pragma unroll 8
        for (int kb = 0; kb < S_ - 4; kb += 4) {
            v2f a_n = *(const v2f*)(Alds + kb + 4);
            v2f b_n = *(const v2f*)(Bv + kb + 4);
            acc = wmma_f32(a_c, b_c, acc);
            a_c = a_n; b_c = b_n;
        }
        acc = wmma_f32(a_c, b_c, acc);
        #pragma unroll
        for (int r = 0; r < 8; ++r) {
            const int q = qbase + rt3 * 16 + r + 8 * half;
            ctx[((size_t)b * S_ + q) * DM_ + h * D_ + ctn + lm] = acc[r];
        }
    }
}

extern "C" void kernel_launch(void* const* d_in, const int* in_sizes, int n_in,
                              void* d_out, int out_size, void* d_ws, size_t ws_size,
                              hipStream_t stream) {
    const float* query          = (const float*)d_in[0];
    const float* key            = (const float*)d_in[1];
    const float* value          = (const float*)d_in[2];
    const unsigned char* q_mask = (const unsigned char*)d_in[3];
    const unsigned char* kv_mask= (const unsigned char*)d_in[4];
    const float* Wq = (const float*)d_in[5];
    const float* bq = (const float*)d_in[6];
    const float* Wk = (const float*)d_in[7];
    const float* bk = (const float*)d_in[8];
    const float* Wv = (const float*)d_in[9];
    const float* bv = (const float*)d_in[10];
    const float* Wo = (const float*)d_in[11];
    const float* bo = (const float*)d_in[12];

    float* out_x = (float*)d_out;                                  // (B,S,DM)
    float* probs = out_x + (size_t)B_ * S_ * DM_;                  // (B,H,S,S)

    const size_t qkv = (size_t)B_ * H_ * S_ * D_;
    float* Qw  = (float*)d_ws;                                     // (B,H,S,D)
    float* Kw  = Qw + qkv;                                         // (B,H,S,D)
    float* Vtw = Kw + qkv;                                         // (B,H,D,S)
    float* ctx = Vtw + qkv;                                        // (B,S,DM)

    dim3 gp(DM_ / 64, (B_ * S_) / 128);
    proj_kernel<<<gp, 256, 0, stream>>>(query, Wq, bq, q_mask,  Qw,  1);
    proj_kernel<<<gp, 256, 0, stream>>>(key,   Wk, bk, kv_mask, Kw,  1);
    proj_kernel<<<gp, 256, 0, stream>>>(value, Wv, bv, kv_mask, Vtw, 2);

    dim3 ga(B_ * H_, S_ / 32);
    attn_kernel<<<ga, 256, 32 * LDSTRIDE * sizeof(float), stream>>>(
        Qw, Kw, Vtw, kv_mask, probs, ctx);

    proj_kernel<<<gp, 256, 0, stream>>>(ctx, Wo, bo, q_mask, out_x, 0);
}